// SpikeDense_5342939316331
// MI455X (gfx1250) — compile-verified
//
#include <hip/hip_runtime.h>

typedef float v2f __attribute__((ext_vector_type(2)));
typedef float v8f __attribute__((ext_vector_type(8)));

#define T_STEPS 8
#define BATCH   16
#define C_IN    256
#define C_OUT   256
#define M_LEN   1024
#define NSTAT   (T_STEPS * BATCH * M_LEN)   // reduction count per channel = 131072

// ---------------------------------------------------------------------------
// Kernel 1: batched GEMM  y[tb, o, m] = sum_c W[o,c] * x[tb, c, m]
// fp32 WMMA 16x16x4. Each wave computes a 32(O) x 128(M) strip:
//   16 accumulators, 2 A fragments + 8 B fragments per k-step -> 16 WMMAs
//   per 18 VMEM ops (~1.1 loads/WMMA). 8 waves/block cover 256(O) x 128(M).
// Also accumulates per-channel sum / sum-of-squares for BatchNorm.
// ---------------------------------------------------------------------------
__global__ __launch_bounds__(256) void snn_gemm(const float* __restrict__ X,
                                                const float* __restrict__ W,
                                                float* __restrict__ Y,
                                                float* __restrict__ csum,
                                                float* __restrict__ csumsq)
{
    const int lane = threadIdx.x & 31;
    const int wave = threadIdx.x >> 5;
    const int tb   = blockIdx.z;            // fused (t,b) batch index
    const int m0   = blockIdx.x * 128;      // M strip origin (shared by block)
    const int o0   = wave * 32;             // O strip origin for this wave
    const int half = lane >> 4;             // 0: lanes 0-15, 1: lanes 16-31
    const int l16  = lane & 15;

    const float* xb    = X + (size_t)tb * C_IN * M_LEN;
    const float* wrow0 = W + (size_t)(o0 + l16) * C_IN;   // A frag, O tile 0
    const float* wrow1 = wrow0 + (size_t)16 * C_IN;       // A frag, O tile 1

    v8f acc[2][8];
#pragma unroll
    for (int ot = 0; ot < 2; ++ot)
#pragma unroll
        for (int mt = 0; mt < 8; ++mt)
            acc[ot][mt] = (v8f){};

    for (int k = 0; k < C_IN; k += 4) {
        // A fragments: lane l -> W[row][k + 2*(l/16)], +1   (8B aligned)
        v2f a0 = *(const v2f*)(wrow0 + k + 2 * half);
        v2f a1 = *(const v2f*)(wrow1 + k + 2 * half);

        // B fragments: lane l -> x[r][col], x[r+1][col], r = k + 2*(l/16)
        const int r = k + 2 * half;
        const float* xr0 = xb + (size_t)r * M_LEN + m0 + l16;
        const float* xr1 = xr0 + M_LEN;
        v2f b[8];
#pragma unroll
        for (int mt = 0; mt < 8; ++mt) {
            b[mt].x = xr0[16 * mt];
            b[mt].y = xr1[16 * mt];
        }

#pragma unroll
        for (int mt = 0; mt < 8; ++mt) {
            acc[0][mt] = __builtin_amdgcn_wmma_f32_16x16x4_f32(
                false, a0, false, b[mt], (short)0, acc[0][mt], false, false);
            acc[1][mt] = __builtin_amdgcn_wmma_f32_16x16x4_f32(
                false, a1, false, b[mt], (short)0, acc[1][mt], false, false);
        }
    }

    // Store y and accumulate per-channel statistics.
    // C/D layout: VGPR j -> output row (tile_o + j + 8*half), column (tile_m + l16).
    float* yb = Y + (size_t)tb * C_OUT * M_LEN;
#pragma unroll
    for (int ot = 0; ot < 2; ++ot) {
#pragma unroll
        for (int j = 0; j < 8; ++j) {
            const int o = o0 + 16 * ot + j + 8 * half;
            float* yo = yb + (size_t)o * M_LEN + m0 + l16;
            float s = 0.0f, sq = 0.0f;
#pragma unroll
            for (int mt = 0; mt < 8; ++mt) {
                float v = acc[ot][mt][j];
                yo[16 * mt] = v;
                s  += v;
                sq += v * v;
            }
            // Reduce across the 16 lanes sharing this channel (xor masks <= 8
            // stay inside each 16-lane half of the wave32).
#pragma unroll
            for (int off = 8; off >= 1; off >>= 1) {
                s  += __shfl_xor(s,  off, 32);
                sq += __shfl_xor(sq, off, 32);
            }
            if (l16 == 0) {
                atomicAdd(&csum[o],   s);
                atomicAdd(&csumsq[o], sq);
            }
        }
    }
}

// ---------------------------------------------------------------------------
// Kernel 2: fold mean/var/gamma/beta into per-channel scale & shift.
// ---------------------------------------------------------------------------
__global__ void snn_stats(const float* __restrict__ csum,
                          const float* __restrict__ csumsq,
                          const float* __restrict__ gamma,
                          const float* __restrict__ beta,
                          float* __restrict__ scale,
                          float* __restrict__ shift)
{
    int o = blockIdx.x * blockDim.x + threadIdx.x;
    if (o < C_OUT) {
        const float inv_n = 1.0f / (float)NSTAT;
        float mean = csum[o] * inv_n;
        float var  = csumsq[o] * inv_n - mean * mean;
        float sc   = gamma[o] * rsqrtf(var + 1e-5f);
        scale[o] = sc;
        shift[o] = beta[o] - mean * sc;
    }
}

// ---------------------------------------------------------------------------
// Kernel 3: BN affine + 8-step LIF recurrence (sequential in t), one thread
// per (b, o, m). v <- 0.5*(v + y); spike at v>=1; hard reset to 0.
// ---------------------------------------------------------------------------
__global__ __launch_bounds__(256) void snn_lif(const float* __restrict__ Y,
                                               const float* __restrict__ scale,
                                               const float* __restrict__ shift,
                                               float* __restrict__ out)
{
    size_t i = (size_t)blockIdx.x * blockDim.x + threadIdx.x;   // over B*C_OUT*M
    const int m = (int)(i & (M_LEN - 1));
    const int o = (int)((i >> 10) & (C_OUT - 1));
    const int b = (int)(i >> 18);

    const float sc = scale[o];
    const float sh = shift[o];
    const size_t base    = ((size_t)b * C_OUT + o) * M_LEN + m;
    const size_t tstride = (size_t)BATCH * C_OUT * M_LEN;

    float v = 0.0f;
#pragma unroll
    for (int t = 0; t < T_STEPS; ++t) {
        float y = Y[base + (size_t)t * tstride] * sc + sh;
        v = 0.5f * (v + y);                       // v += (y - v)/tau, tau=2
        float s = (v >= 1.0f) ? 1.0f : 0.0f;      // Heaviside(v - VTH)
        out[base + (size_t)t * tstride] = s;
        v = (s > 0.0f) ? 0.0f : v;                // hard reset
    }
}

extern "C" void kernel_launch(void* const* d_in, const int* in_sizes, int n_in,
                              void* d_out, int out_size, void* d_ws, size_t ws_size,
                              hipStream_t stream) {
    const float* x     = (const float*)d_in[0];  // [8,16,256,1024]
    const float* W     = (const float*)d_in[1];  // [256,256]
    const float* gamma = (const float*)d_in[2];  // [256]
    const float* beta  = (const float*)d_in[3];  // [256]
    float* out = (float*)d_out;                  // [8,16,256,1024]

    // Workspace layout: y intermediate (134 MB), then small stat buffers.
    const size_t ybytes = (size_t)T_STEPS * BATCH * C_OUT * M_LEN * sizeof(float);
    float* Y      = (float*)d_ws;
    float* csum   = (float*)((char*)d_ws + ybytes);
    float* csumsq = csum + C_OUT;
    float* scale  = csumsq + C_OUT;
    float* shift  = scale + C_OUT;

    // Zero the atomic accumulators (graph-capture-safe async memset).
    hipMemsetAsync(csum, 0, 2 * C_OUT * sizeof(float), stream);

    dim3 g(M_LEN / 128, 1, T_STEPS * BATCH);            // 8 x 1 x 128 blocks
    snn_gemm<<<g, 256, 0, stream>>>(x, W, Y, csum, csumsq);

    snn_stats<<<1, 256, 0, stream>>>(csum, csumsq, gamma, beta, scale, shift);

    const size_t n = (size_t)BATCH * C_OUT * M_LEN;     // 4,194,304 threads
    snn_lif<<<(unsigned)(n / 256), 256, 0, stream>>>(Y, scale, shift, out);
}